// DistMul_23536420782557
// MI455X (gfx1250) — compile-verified
//
#include <hip/hip_runtime.h>
#include <math.h>

// DistMult scoring for MI455X (gfx1250, wave32).
// score_b = sigmoid( sum_d h[b][d] * r[b][d] * t[b][d] ),  d = 128.
//
// Memory-bound (~268 MB of random gathers -> ~11.5 us @ 23.3 TB/s).
// Strategy: per wave, tile of 16 batch elements. Coalesced b128 gathers of
// h/t/r rows -> P = h*t and R staged in LDS (row stride 132 floats => every
// ds_load_b64 in the WMMA feed hits 64 distinct banks). Dot products done
// with chained V_WMMA_F32_16X16X4_F32 (full f32 precision, matches the f32
// reference); the diagonal of D = P x R^T is the score vector.

typedef __attribute__((ext_vector_type(2))) float v2f;
typedef __attribute__((ext_vector_type(8))) float v8f;

#define WAVES_PER_BLOCK 8
#define ROWS 16              // batch elements per wave (WMMA M = N = 16)
#define DIM 128
#define LDS_STRIDE 132       // 128 + 4 pad: stride == 4 (mod 64) -> conflict-free
#define TILE (ROWS * LDS_STRIDE)

__global__ void distmult_wmma_kernel(const int* __restrict__ bh,
                                     const int* __restrict__ bt,
                                     const int* __restrict__ br,
                                     const float* __restrict__ ent,
                                     const float* __restrict__ rel,
                                     float* __restrict__ out,
                                     int batch) {
  extern __shared__ float lds[];
  const int lane = threadIdx.x & 31;
  const int wave = threadIdx.x >> 5;
  float* P  = lds + wave * (2 * TILE);   // h*t products: 16 rows x 128 (stride 132)
  float* Rt = P + TILE;                  // relation rows: 16 rows x 128 (stride 132)

  const int tileBase = (blockIdx.x * WAVES_PER_BLOCK + wave) * ROWS;
  if (tileBase >= batch) return;        // wave-uniform; EXEC stays all-1 for WMMA

  // ---- Stage: coalesced gather of 16 (h,t,r) rows.
  // One global_load_b128 per row per tensor covers the full 512B row across
  // the 32 lanes. Indices are wave-uniform -> scalar loads (KMcnt path).
#pragma unroll
  for (int r = 0; r < ROWS; ++r) {
    const int b = tileBase + r;
    const long hi = bh[b];
    const long ti = bt[b];
    const long ri = br[b];
    const float4 hv = *(const float4*)(ent + hi * DIM + lane * 4);
    const float4 tv = *(const float4*)(ent + ti * DIM + lane * 4);
    const float4 rv = *(const float4*)(rel + ri * DIM + lane * 4);
    float4 pv;
    pv.x = hv.x * tv.x;
    pv.y = hv.y * tv.y;
    pv.z = hv.z * tv.z;
    pv.w = hv.w * tv.w;
    *(float4*)(P  + r * LDS_STRIDE + lane * 4) = pv;
    *(float4*)(Rt + r * LDS_STRIDE + lane * 4) = rv;
  }

  // ---- Compute: D += A x B over K = 128 in chunks of 4 (f32 WMMA).
  // f32 16x4 A layout: lanes 0-15 hold (row, K=0..1), lanes 16-31 (row, K=2..3).
  // Required B layout (B[k][j] = R[j][4c+k]) has the *same* per-lane addressing,
  // so both operands are a single ds_load_b64 per chunk.
  const int row  = lane & 15;
  const int koff = (lane >> 4) << 1;          // 0 for lanes 0-15, 2 for 16-31
  const int base = row * LDS_STRIDE + koff;
  v8f acc = {0.f, 0.f, 0.f, 0.f, 0.f, 0.f, 0.f, 0.f};
#pragma unroll
  for (int c = 0; c < DIM / 4; ++c) {
    v2f a = *(const v2f*)(P  + base + c * 4);
    v2f b = *(const v2f*)(Rt + base + c * 4);
    // (neg_a, A, neg_b, B, c_mod, C, reuse_a, reuse_b)
    acc = __builtin_amdgcn_wmma_f32_16x16x4_f32(false, a, false, b,
                                                (short)0, acc, false, false);
  }

  // ---- Diagonal extract (D[i][i] = score_i), sigmoid, store.
  // C/D layout: VGPR v, lanes 0-15 -> M=v, N=lane; lanes 16-31 -> M=v+8, N=lane-16.
  float s = 0.f;
  int oidx = -1;
#pragma unroll
  for (int v = 0; v < 8; ++v) {
    if (lane == v)      { s = acc[v]; oidx = v; }       // (v, v)      @ lane v
    if (lane == v + 24) { s = acc[v]; oidx = v + 8; }   // (v+8, v+8)  @ lane v+24
  }
  if (oidx >= 0) {
    const int o = tileBase + oidx;
    if (o < batch) out[o] = 1.f / (1.f + __expf(-s));
  }
}

extern "C" void kernel_launch(void* const* d_in, const int* in_sizes, int n_in,
                              void* d_out, int out_size, void* d_ws, size_t ws_size,
                              hipStream_t stream) {
  const int* bh = (const int*)d_in[0];
  const int* bt = (const int*)d_in[1];
  const int* br = (const int*)d_in[2];
  const float* ent = (const float*)d_in[3];
  const float* rel = (const float*)d_in[4];
  float* out = (float*)d_out;

  const int batch = in_sizes[0];
  const int elemsPerBlock = WAVES_PER_BLOCK * ROWS;          // 128
  const int blocks = (batch + elemsPerBlock - 1) / elemsPerBlock;
  const size_t shmem = (size_t)WAVES_PER_BLOCK * 2 * TILE * sizeof(float); // 132 KB

  distmult_wmma_kernel<<<dim3(blocks), dim3(WAVES_PER_BLOCK * 32), shmem, stream>>>(
      bh, bt, br, ent, rel, out, batch);
}